// GNNSafe_88682484728259
// MI455X (gfx1250) — compile-verified
//
#include <hip/hip_runtime.h>

// GNNSafe propagation: e <- 0.5*e + 0.5 * (D^-1-normalized scatter-sum), 2 layers.
// Memory-bound sparse op. Edge indices (128MB int32) fit MI455X's 192MB L2, so
// repeated passes are L2-resident; node vectors (2MB) live in L2 throughout.
// CDNA5 path: wave-private double-buffered global_load_async_to_lds_b128 staging
// of edge-index tiles, pipelined with s_wait_asynccnt / s_wait_dscnt.
// NOTE: k_scatter is defined FIRST so the disasm snippet shows the hot kernel.

#define WAVE_SZ 32
#define EPT 4                       // 4 int32 edges per lane per tile (one b128)
#define TILE (WAVE_SZ * EPT)        // 128 edges per wave-tile
#define WAVES_PER_BLOCK 8
#define BLOCK (WAVES_PER_BLOCK * WAVE_SZ)   // 256 threads = 8 wave32
#define ALPHA_F 0.5f

#if defined(__has_builtin)
#  if __has_builtin(__builtin_amdgcn_global_load_async_to_lds_b128)
#    define HAVE_ASYNC_LDS 1
#  endif
#endif

#ifdef HAVE_ASYNC_LDS
typedef int v4i __attribute__((vector_size(16)));
typedef __attribute__((address_space(1))) v4i* gv4i_ptr;
typedef __attribute__((address_space(3))) v4i* lv4i_ptr;

__device__ __forceinline__ void async_ld_b128(const int* g, int* l) {
  __builtin_amdgcn_global_load_async_to_lds_b128(
      (gv4i_ptr)(v4i*)(int*)g,
      (lv4i_ptr)(v4i*)l,
      /*offset=*/0, /*cpol=*/0);
}
#endif

// acc[col[k]] += e_src[row[k]] over full 128-edge wave tiles.
// Per-wave double-buffered LDS staging of the index streams via async copies.
__global__ void __launch_bounds__(BLOCK)
k_scatter(const int* __restrict__ row, const int* __restrict__ col,
          const float* __restrict__ e_src, float* __restrict__ acc, int ntiles) {
#ifdef HAVE_ASYNC_LDS
  __shared__ int s_row[WAVES_PER_BLOCK][2][TILE];
  __shared__ int s_col[WAVES_PER_BLOCK][2][TILE];
  const int w     = threadIdx.x >> 5;
  const int lane  = threadIdx.x & 31;
  const int lbase = lane * EPT;
  const int gw    = blockIdx.x * WAVES_PER_BLOCK + w;
  const int nw    = gridDim.x * WAVES_PER_BLOCK;

  int t = gw;
  if (t < ntiles) {  // prologue: stage first tile into buffer 0 (ASYNCcnt += 2)
    async_ld_b128(row + (size_t)t * TILE + lbase, &s_row[w][0][lbase]);
    async_ld_b128(col + (size_t)t * TILE + lbase, &s_col[w][0][lbase]);
  }
  int cur = 0;
  for (; t < ntiles; t += nw) {
    int tn = t + nw;
    if (tn < ntiles) {
      // WAR guard: prior LDS reads of buffer cur^1 must retire before overwrite.
      asm volatile("s_wait_dscnt 0x0" ::: "memory");
      async_ld_b128(row + (size_t)tn * TILE + lbase, &s_row[w][cur ^ 1][lbase]);
      async_ld_b128(col + (size_t)tn * TILE + lbase, &s_col[w][cur ^ 1][lbase]);
      // Async loads retire in order: cnt<=2 means current tile's 2 loads landed.
      asm volatile("s_wait_asynccnt 0x2" ::: "memory");
    } else {
      asm volatile("s_wait_asynccnt 0x0" ::: "memory");
    }
#pragma unroll
    for (int j = 0; j < EPT; ++j) {
      int r = s_row[w][cur][lbase + j];
      int c = s_col[w][cur][lbase + j];
      unsafeAtomicAdd(&acc[c], e_src[r]);   // both L2-resident (2MB node arrays)
    }
    cur ^= 1;
  }
#else
  // Fallback: direct b128 index loads, grid-stride.
  int e4 = ntiles * (TILE / 4);
  int stride = gridDim.x * blockDim.x;
  for (int i = blockIdx.x * blockDim.x + threadIdx.x; i < e4; i += stride) {
    int4 r4 = reinterpret_cast<const int4*>(row)[i];
    int4 c4 = reinterpret_cast<const int4*>(col)[i];
    unsafeAtomicAdd(&acc[c4.x], e_src[r4.x]);
    unsafeAtomicAdd(&acc[c4.y], e_src[r4.y]);
    unsafeAtomicAdd(&acc[c4.z], e_src[r4.z]);
    unsafeAtomicAdd(&acc[c4.w], e_src[r4.w]);
  }
#endif
}

__global__ void k_zero2(float* __restrict__ a, float* __restrict__ b, int n) {
  int i = blockIdx.x * blockDim.x + threadIdx.x;
  if (i < n) { a[i] = 0.0f; b[i] = 0.0f; }
}

// In-degree histogram: deg[col[k]] += 1.0 (float is exact for counts < 2^24).
__global__ void k_deg4(const int* __restrict__ col, float* __restrict__ deg, int e4) {
  int i = blockIdx.x * blockDim.x + threadIdx.x;
  if (i < e4) {
    int4 c = reinterpret_cast<const int4*>(col)[i];
    unsafeAtomicAdd(&deg[c.x], 1.0f);
    unsafeAtomicAdd(&deg[c.y], 1.0f);
    unsafeAtomicAdd(&deg[c.z], 1.0f);
    unsafeAtomicAdd(&deg[c.w], 1.0f);
  }
}

__global__ void k_deg_tail(const int* __restrict__ col, float* __restrict__ deg,
                           int start, int E) {
  int i = start + blockIdx.x * blockDim.x + threadIdx.x;
  if (i < E) unsafeAtomicAdd(&deg[col[i]], 1.0f);
}

__global__ void k_invdeg(float* __restrict__ deg, int n) {
  int i = blockIdx.x * blockDim.x + threadIdx.x;
  if (i < n) {
    float d = deg[i];
    deg[i] = (d > 0.0f) ? (1.0f / d) : 0.0f;
  }
}

__global__ void k_scatter_tail(const int* __restrict__ row, const int* __restrict__ col,
                               const float* __restrict__ e_src, float* __restrict__ acc,
                               int start, int E) {
  int i = start + blockIdx.x * blockDim.x + threadIdx.x;
  if (i < E) unsafeAtomicAdd(&acc[col[i]], e_src[row[i]]);
}

// e_dst = alpha*e_src + (1-alpha) * inv_deg * acc ; optionally re-zero acc.
__global__ void k_blend(const float* __restrict__ e_src, float* __restrict__ acc,
                        const float* __restrict__ invdeg, float* __restrict__ e_dst,
                        int n, int clear_acc) {
  int i = blockIdx.x * blockDim.x + threadIdx.x;
  if (i < n) {
    float m = invdeg[i] * acc[i];
    e_dst[i] = ALPHA_F * e_src[i] + (1.0f - ALPHA_F) * m;
    if (clear_acc) acc[i] = 0.0f;
  }
}

extern "C" void kernel_launch(void* const* d_in, const int* in_sizes, int n_in,
                              void* d_out, int out_size, void* d_ws, size_t ws_size,
                              hipStream_t stream) {
  const float* e    = (const float*)d_in[0];
  const int*   edge = (const int*)d_in[1];       // int32 (JAX default x64 off)
  const int N = in_sizes[0];
  const int E = in_sizes[1] / 2;
  const int* rowp = edge;
  const int* colp = edge + (size_t)E;

  float* deg = (float*)d_ws;        // N floats; becomes inv_deg in place
  float* acc = deg + N;             // N floats: scatter accumulator
  float* e1  = acc + N;             // N floats: after layer 1
  float* out = (float*)d_out;

  const int nb_nodes = (N + BLOCK - 1) / BLOCK;

  // 0) zero accumulators every call (graph-replay safe)
  k_zero2<<<nb_nodes, BLOCK, 0, stream>>>(deg, acc, N);

  // 1) in-degree
  const int e4 = E / 4;
  if (e4 > 0) {
    k_deg4<<<(e4 + BLOCK - 1) / BLOCK, BLOCK, 0, stream>>>(colp, deg, e4);
  }
  if (e4 * 4 < E) {
    k_deg_tail<<<1, 256, 0, stream>>>(colp, deg, e4 * 4, E);
  }

  // 2) inv_deg = deg>0 ? 1/deg : 0 (in place)
  k_invdeg<<<nb_nodes, BLOCK, 0, stream>>>(deg, N);

  const int ntiles = E / TILE;
  const int tiled  = ntiles * TILE;
  int nb_sc = (ntiles + WAVES_PER_BLOCK - 1) / WAVES_PER_BLOCK;
  if (nb_sc > 2048) nb_sc = 2048;   // ~16K waves, ~8 tiles each: deep async pipeline

  // 3) layer 1: acc = scatter(e); e1 = 0.5 e + 0.5 inv_deg*acc; acc = 0
  if (ntiles > 0)
    k_scatter<<<nb_sc, BLOCK, 0, stream>>>(rowp, colp, e, acc, ntiles);
  if (tiled < E)
    k_scatter_tail<<<(E - tiled + 255) / 256, 256, 0, stream>>>(rowp, colp, e, acc, tiled, E);
  k_blend<<<nb_nodes, BLOCK, 0, stream>>>(e, acc, deg, e1, N, 1);

  // 4) layer 2 -> d_out
  if (ntiles > 0)
    k_scatter<<<nb_sc, BLOCK, 0, stream>>>(rowp, colp, e1, acc, ntiles);
  if (tiled < E)
    k_scatter_tail<<<(E - tiled + 255) / 256, 256, 0, stream>>>(rowp, colp, e1, acc, tiled, E);
  k_blend<<<nb_nodes, BLOCK, 0, stream>>>(e1, acc, deg, out, N, 0);
}